// CompiledCombinedLogicNet_6932077216082
// MI455X (gfx1250) — compile-verified
//
#include <hip/hip_runtime.h>
#include <stdint.h>
#include <stddef.h>

// Problem constants (match reference)
constexpr int B = 16384;   // batch
constexpr int W = 16384;   // wires per layer
constexpr int L = 6;       // layers
constexpr int C = 16;      // classes
constexpr int BDIM = 512;  // 16 wave32 waves per workgroup

// ---- CDNA5 TDM availability probe -----------------------------------------
#if defined(__has_builtin)
#if __has_builtin(__builtin_amdgcn_tensor_load_to_lds) && \
    __has_builtin(__builtin_amdgcn_s_wait_tensorcnt)
#define USE_TDM 1
#endif
#endif
#ifndef USE_TDM
#define USE_TDM 0
#endif

typedef uint32_t u32x4 __attribute__((ext_vector_type(4)));
typedef int      i32x4 __attribute__((ext_vector_type(4)));
typedef int      i32x8 __attribute__((ext_vector_type(8)));

typedef __attribute__((address_space(3))) uint8_t lds_byte_t;

// Byte offset of a generic (__shared__) pointer within LDS (AS3 is 32-bit).
__device__ __forceinline__ uint32_t lds_off(void* p) {
  return (uint32_t)(uintptr_t)(lds_byte_t*)p;
}

// Issue one TDM DMA: copy W uint32 gate descriptors (64 KB, contiguous) from
// global memory into LDS at byte offset `ldsoff`.  D# built per CDNA5 ISA §8.
// This toolchain's builtin takes 6 args:
//   (uint32x4 group0, int32x8 group1, int32x4 group2, int32x4 group3,
//    int32x8 extra, int cpol)
__device__ __forceinline__ void tdm_issue(const uint32_t* gsrc, uint32_t ldsoff) {
#if USE_TDM
  uint64_t ga = (uint64_t)(uintptr_t)gsrc;
  // Group 0: count=1 | lds_addr | global_addr[56:0] | type=2 ("image")
  u32x4 g0 = { 1u,
               ldsoff,
               (uint32_t)ga,
               (uint32_t)((ga >> 32) & 0x1FFFFFFull) | (2u << 30) };
  // Group 1: data_size=4B; tensor_dim0=W; tensor_dim1=1; tile_dim0=W;
  //          tile_dim1=1; tensor_dim0_stride=W  (1 row of W dwords)
  i32x8 g1 = { (int)(2u << 16),          // workgroup_mask=0, data_size=2 (4B)
               (int)((uint32_t)W << 16), // tensor_dim0[15:0] in [31:16]
               (int)(1u << 16),          // tensor_dim0 hi=0; tensor_dim1 lo=1
               (int)((uint32_t)W << 16), // tensor_dim1 hi=0; tile_dim0=W
               1,                        // tile_dim1=1, tile_dim2=0
               W,                        // tensor_dim0_stride[31:0]
               0, 0 };
  i32x4 gz4 = { 0, 0, 0, 0 };            // groups 2/3 unused (<=2D tensor)
  i32x8 gz8 = { 0, 0, 0, 0, 0, 0, 0, 0 };
  __builtin_amdgcn_tensor_load_to_lds(g0, g1, gz4, gz4, gz8, 0);
#endif
}

// One gate: bit-sliced 4-entry truth table over 32 batch elements.
// op bit0 -> (a,b)=(1,1); bit1 -> (1,0); bit2 -> (0,1); bit3 -> (0,0)
__device__ __forceinline__ uint32_t gate_eval(uint32_t de,
                                              const uint32_t* __restrict__ src) {
  uint32_t a  = src[de & 0x3FFFu];
  uint32_t b  = src[(de >> 14) & 0x3FFFu];
  uint32_t op = de >> 28;
  return ((a & b)  & (0u - (op & 1u)))
       | ((a & ~b) & (0u - ((op >> 1) & 1u)))
       | ((~a & b) & (0u - ((op >> 2) & 1u)))
       | (~(a | b) & (0u - ((op >> 3) & 1u)));
}

// ---- Kernel 1: pack (idx_a, idx_b, op) -> one uint32 descriptor -----------
__global__ void pack_desc_kernel(const int* __restrict__ ia,
                                 const int* __restrict__ ib,
                                 const int* __restrict__ op,
                                 uint32_t* __restrict__ desc, int n) {
  int i = blockIdx.x * blockDim.x + threadIdx.x;
  if (i < n) {
    desc[i] = ((uint32_t)ia[i] & 0x3FFFu) |
              (((uint32_t)ib[i] & 0x3FFFu) << 14) |
              (((uint32_t)op[i] & 0xFu) << 28);
  }
}

// ---- Kernel 2: bit-sliced logic net, one workgroup per 32 batch rows ------
// Dynamic LDS layout (uint32 words):
//   [0      , W)   state buffer 0
//   [W      , 2W)  state buffer 1
//   [2W     , 3W)  descriptor buffer 0
//   [3W     , 4W)  descriptor buffer 1
__global__ __launch_bounds__(BDIM) void logic_net_kernel(
    const float* __restrict__ x, const uint32_t* __restrict__ desc,
    float* __restrict__ out) {
  extern __shared__ uint32_t smem[];
  uint32_t* st0 = smem;
  uint32_t* st1 = smem + W;

  const int t = threadIdx.x;
  const int g = blockIdx.x;  // batch group: rows [32g, 32g+32)

  // Prefetch layer-0 descriptors while we bitpack the inputs.
#if USE_TDM
  if (t < 32) tdm_issue(desc, lds_off(smem + 2 * W));
#else
  for (int i = t; i < W; i += BDIM) smem[2 * W + i] = desc[i];
#endif

  // Threshold + bitpack with 128-bit streaming loads:
  // word[w] bit j = (x[32g+j, w] > 0.5).  Each thread builds 4 adjacent words.
  {
    const float4* base4 = (const float4*)(x + (size_t)g * 32u * (size_t)W);
    for (int q = t; q < W / 4; q += BDIM) {
      uint32_t w0 = 0u, w1 = 0u, w2 = 0u, w3 = 0u;
#pragma unroll
      for (int j = 0; j < 32; ++j) {
        float4 v = base4[(size_t)j * (W / 4) + q];
        uint32_t bit = 1u << j;
        w0 |= (v.x > 0.5f) ? bit : 0u;
        w1 |= (v.y > 0.5f) ? bit : 0u;
        w2 |= (v.z > 0.5f) ? bit : 0u;
        w3 |= (v.w > 0.5f) ? bit : 0u;
      }
      ((uint4*)st0)[q] = make_uint4(w0, w1, w2, w3);
    }
  }
  __syncthreads();

  // Gate layers: 32 batches per word, pure bitwise truth-table evaluation.
  for (int l = 0; l < L; ++l) {
    uint32_t* db_cur = smem + 2 * W + (size_t)(l & 1) * W;
    uint32_t* db_nxt = smem + 2 * W + (size_t)((l + 1) & 1) * W;
#if USE_TDM
    if (t < 32) __builtin_amdgcn_s_wait_tensorcnt(0);  // db_cur ready
    __syncthreads();
    if ((l + 1 < L) && (t < 32))
      tdm_issue(desc + (size_t)(l + 1) * W, lds_off(db_nxt));
#else
    if (l + 1 < L)
      for (int i = t; i < W; i += BDIM)
        db_nxt[i] = desc[(size_t)(l + 1) * W + i];
#endif
    const uint32_t* __restrict__ src = (l & 1) ? st1 : st0;
    const uint4* __restrict__ dl4    = (const uint4*)db_cur;
    uint4* __restrict__ dst4         = (uint4*)((l & 1) ? st0 : st1);
    for (int q = t; q < W / 4; q += BDIM) {
      uint4 de = dl4[q];                       // 4 packed gate descriptors
      uint32_t r0 = gate_eval(de.x, src);      // 4 independent LDS gather pairs
      uint32_t r1 = gate_eval(de.y, src);
      uint32_t r2 = gate_eval(de.z, src);
      uint32_t r3 = gate_eval(de.w, src);
      dst4[q] = make_uint4(r0, r1, r2, r3);
    }
    __syncthreads();
  }

  // GroupSum: wave = class, lane = batch bit.  L=6 even -> result in st0.
  {
    const int wv = t >> 5;   // class 0..15
    const int ln = t & 31;   // batch bit within group
    const uint4* hc4 = (const uint4*)(st0 + wv * (W / C));
    uint32_t cnt = 0;
    for (int k = 0; k < (W / C) / 4; ++k) {
      uint4 v = hc4[k];                        // broadcast b128 LDS load
      cnt += (v.x >> ln) & 1u;
      cnt += (v.y >> ln) & 1u;
      cnt += (v.z >> ln) & 1u;
      cnt += (v.w >> ln) & 1u;
    }
    out[((size_t)g * 32 + ln) * (size_t)C + wv] = (float)cnt;
  }
}

extern "C" void kernel_launch(void* const* d_in, const int* in_sizes, int n_in,
                              void* d_out, int out_size, void* d_ws, size_t ws_size,
                              hipStream_t stream) {
  (void)in_sizes; (void)n_in; (void)out_size; (void)ws_size;
  const float* x  = (const float*)d_in[0];
  const int*   ia = (const int*)d_in[1];
  const int*   ib = (const int*)d_in[2];
  const int*   op = (const int*)d_in[3];
  float*       out  = (float*)d_out;
  uint32_t*    desc = (uint32_t*)d_ws;  // L*W*4 = 384 KB of workspace

  const int n = L * W;
  pack_desc_kernel<<<(n + 255) / 256, 256, 0, stream>>>(ia, ib, op, desc, n);

  const size_t shmem = (size_t)4 * W * sizeof(uint32_t);  // 256 KB
  logic_net_kernel<<<B / 32, BDIM, shmem, stream>>>(x, desc, out);
}